// ODST_17669495456365
// MI455X (gfx1250) — compile-verified
//
#include <hip/hip_runtime.h>
#include <hip/hip_bf16.h>

// CDNA5 / gfx1250, wave32. f32 WMMA fragments:
//  A (16x4 f32): 2 VGPRs/lane. lane l: M = l%16, holds K = (l/16)*2 + {0,1}
//  B (4x16 f32): 2 VGPRs/lane. lane l: N = l%16, holds K = (l/16)*2 + {0,1}
//  C/D (16x16 f32): 8 VGPRs/lane. VGPR r: M = r + (l/16)*8, N = l%16
typedef __attribute__((ext_vector_type(2))) float v2f;
typedef __attribute__((ext_vector_type(8))) float v8f;

#define B_DIM 1024
#define I_DIM 256
#define N_DIM 512
#define D_DIM 6
#define U_DIM 16
#define ND    (N_DIM * D_DIM)      // 3072
#define NLEAF 64                   // 2^D

// ---------------------------------------------------------------------------
// Kernel 1: sparsemax over D=6 per (i,n); write selectors transposed
// selT[col = n*6+d][i], layout [ND][I] so GEMM B-columns are contiguous in K.
// ---------------------------------------------------------------------------
__global__ __launch_bounds__(256) void k_sparsemax(
    const float* __restrict__ fsl,   // [I, N, D]
    float* __restrict__ selT)        // [ND, I]
{
  int r = blockIdx.x * blockDim.x + threadIdx.x;   // 0 .. I*N-1
  if (r >= I_DIM * N_DIM) return;
  int i = r >> 9;          // r / N_DIM
  int n = r & (N_DIM - 1); // r % N_DIM

  float z[D_DIM];
  const float* src = fsl + r * D_DIM;
  #pragma unroll
  for (int d = 0; d < D_DIM; ++d) z[d] = src[d];

  // descending sort of a copy, optimal 12-comparator network for 6 elems
  float s0 = z[0], s1 = z[1], s2 = z[2], s3 = z[3], s4 = z[4], s5 = z[5];
  #define CSWAP(a, b) { float hi = fmaxf(a, b), lo = fminf(a, b); a = hi; b = lo; }
  CSWAP(s0, s1) CSWAP(s2, s3) CSWAP(s4, s5)
  CSWAP(s0, s2) CSWAP(s3, s5) CSWAP(s1, s4)
  CSWAP(s0, s1) CSWAP(s2, s3) CSWAP(s4, s5)
  CSWAP(s1, s2) CSWAP(s3, s4) CSWAP(s2, s3)
  #undef CSWAP
  float srt[D_DIM] = { s0, s1, s2, s3, s4, s5 };

  float csum = 0.0f, tausum = 0.0f;
  int kz = 1;
  #pragma unroll
  for (int k = 0; k < D_DIM; ++k) {
    csum += srt[k];
    if (1.0f + (float)(k + 1) * srt[k] > csum) { kz = k + 1; tausum = csum; }
  }
  float tau = (tausum - 1.0f) / (float)kz;

  #pragma unroll
  for (int d = 0; d < D_DIM; ++d)
    selT[(n * D_DIM + d) * I_DIM + i] = fmaxf(z[d] - tau, 0.0f);
}

// ---------------------------------------------------------------------------
// Kernel 2: feature_values = x @ selectors via V_WMMA_F32_16X16X4_F32.
// Register-tiled 2x2: one wave owns a 32x32 output block (4 accumulators),
// so each A/B fragment load feeds two WMMAs -> halves L2 traffic vs 16x16.
// Fused sparsemoid epilogue: s = clip(0.5 + 0.5*(fv - th)*exp(-lt)).
// ---------------------------------------------------------------------------
__global__ __launch_bounds__(128) void k_gemm1(
    const float* __restrict__ x,     // [B, I]
    const float* __restrict__ selT,  // [ND, I]
    const float* __restrict__ th,    // [N, D] flat [ND]
    const float* __restrict__ lt,    // [N, D] flat [ND]
    float* __restrict__ fvs)         // [B, ND]
{
  const int lane  = threadIdx.x & 31;
  const int wave  = threadIdx.x >> 5;
  const int w     = blockIdx.x * 4 + wave;        // 0..3071 (32x32 tiles)
  const int rowT  = w / (ND / 32);                // 0..31  batch tiles of 32
  const int colT  = w % (ND / 32);                // 0..95  nd tiles of 32
  const int m     = lane & 15;
  const int khalf = lane >> 4;                    // 0/1 -> K offset 0/2

  const float* aPtr0 = x    + (rowT * 32 + m) * I_DIM + khalf * 2;
  const float* aPtr1 = aPtr0 + 16 * I_DIM;
  const float* bPtr0 = selT + (colT * 32 + m) * I_DIM + khalf * 2;
  const float* bPtr1 = bPtr0 + 16 * I_DIM;

  v8f acc00 = {}, acc01 = {}, acc10 = {}, acc11 = {};
  #pragma unroll 4
  for (int k0 = 0; k0 < I_DIM; k0 += 4) {
    v2f a0 = *(const v2f*)(aPtr0 + k0);
    v2f a1 = *(const v2f*)(aPtr1 + k0);
    v2f b0 = *(const v2f*)(bPtr0 + k0);
    v2f b1 = *(const v2f*)(bPtr1 + k0);
    acc00 = __builtin_amdgcn_wmma_f32_16x16x4_f32(false, a0, false, b0, (short)0, acc00, false, false);
    acc01 = __builtin_amdgcn_wmma_f32_16x16x4_f32(false, a0, false, b1, (short)0, acc01, false, false);
    acc10 = __builtin_amdgcn_wmma_f32_16x16x4_f32(false, a1, false, b0, (short)0, acc10, false, false);
    acc11 = __builtin_amdgcn_wmma_f32_16x16x4_f32(false, a1, false, b1, (short)0, acc11, false, false);
  }

  // Epilogue: each lane owns two columns (n,d) and two row groups of 8.
  const int col0 = colT * 32 + m;
  const int col1 = col0 + 16;
  const float t0 = th[col0], i0 = __expf(-lt[col0]);
  const float t1 = th[col1], i1 = __expf(-lt[col1]);
  const int rb0  = rowT * 32 + khalf * 8;
  const int rb1  = rb0 + 16;
  #pragma unroll
  for (int r = 0; r < 8; ++r) {
    float v;
    v = (acc00[r] - t0) * i0;
    fvs[(rb0 + r) * ND + col0] = fminf(fmaxf(0.5f * v + 0.5f, 0.0f), 1.0f);
    v = (acc01[r] - t1) * i1;
    fvs[(rb0 + r) * ND + col1] = fminf(fmaxf(0.5f * v + 0.5f, 0.0f), 1.0f);
    v = (acc10[r] - t0) * i0;
    fvs[(rb1 + r) * ND + col0] = fminf(fmaxf(0.5f * v + 0.5f, 0.0f), 1.0f);
    v = (acc11[r] - t1) * i1;
    fvs[(rb1 + r) * ND + col1] = fminf(fmaxf(0.5f * v + 0.5f, 0.0f), 1.0f);
  }
}

// ---------------------------------------------------------------------------
// Kernel 3: leaf weights + response GEMM.
// rw[b,n,c] = prod_d ( bit_d(c) ? 1-s[b,n,d] : s[b,n,d] )
// out[b,u] += sum_n rw[b,n,:] . response[n,u,:]   (K = 64, via 16 WMMAs/n)
// One wave per (batch tile, n-chunk of 32); partials merged with f32 atomics.
// ---------------------------------------------------------------------------
__global__ __launch_bounds__(32) void k_leafgemm(
    const float* __restrict__ fvs,   // [B, ND]
    const float* __restrict__ resp,  // [N, U, 64]
    float* __restrict__ out)         // [B, U], pre-zeroed
{
  __shared__ float sld[16 * D_DIM]; // s values for 16 batches of one tree
  const int lane  = threadIdx.x;
  const int bt    = blockIdx.x;     // 0..63  batch tile
  const int chunk = blockIdx.y;     // 0..15  n chunk
  const int m     = lane & 15;      // batch row (A) / unit (B)
  const int khalf = lane >> 4;      // K offset half

  v8f acc = {};
  for (int nn = 0; nn < 32; ++nn) {
    const int n = chunk * 32 + nn;
    // stage s[16][6] for this tree into LDS (3 loads/lane)
    #pragma unroll
    for (int t = lane; t < 16 * D_DIM; t += 32)
      sld[t] = fvs[(bt * 16 + t / D_DIM) * ND + n * D_DIM + (t % D_DIM)];
    __syncthreads();

    const float s0 = sld[m * D_DIM + 0];
    const float s1 = sld[m * D_DIM + 1];
    const float s2 = sld[m * D_DIM + 2];
    const float s3 = sld[m * D_DIM + 3];
    const float s4 = sld[m * D_DIM + 4];
    const float s5 = sld[m * D_DIM + 5];
    // leaf index for this lane at kstep: c = kstep*4 + khalf*2 + {0,1}
    // bit0 = {0,1}, bit1 = khalf, bits2..5 = kstep bits (compile-time by unroll)
    const float t1 = khalf ? (1.0f - s1) : s1;

    const float* bBase = resp + (n * U_DIM + m) * NLEAF + khalf * 2;
    #pragma unroll
    for (int kstep = 0; kstep < 16; ++kstep) {
      float p = t1;
      p *= (kstep & 1)        ? (1.0f - s2) : s2;
      p *= ((kstep >> 1) & 1) ? (1.0f - s3) : s3;
      p *= ((kstep >> 2) & 1) ? (1.0f - s4) : s4;
      p *= ((kstep >> 3) & 1) ? (1.0f - s5) : s5;
      v2f a; a.x = p * s0; a.y = p * (1.0f - s0);
      v2f b = *(const v2f*)(bBase + kstep * 4);
      acc = __builtin_amdgcn_wmma_f32_16x16x4_f32(
          false, a, false, b, (short)0, acc, false, false);
    }
    __syncthreads();
  }

  // merge partial 16x16 tile (16 batches x 16 units) into out
  const int rb = bt * 16 + khalf * 8;
  #pragma unroll
  for (int r = 0; r < 8; ++r)
    atomicAdd(&out[(rb + r) * U_DIM + m], acc[r]);
}

// ---------------------------------------------------------------------------
extern "C" void kernel_launch(void* const* d_in, const int* in_sizes, int n_in,
                              void* d_out, int out_size, void* d_ws, size_t ws_size,
                              hipStream_t stream) {
  const float* x   = (const float*)d_in[0];   // [1024, 256]
  const float* fsl = (const float*)d_in[1];   // [256, 512, 6]
  const float* th  = (const float*)d_in[2];   // [512, 6]
  const float* lt  = (const float*)d_in[3];   // [512, 6]
  const float* rsp = (const float*)d_in[4];   // [512, 16, 64]
  float* out = (float*)d_out;                 // [1024, 16]

  // workspace layout
  float* selT = (float*)d_ws;                       // [3072, 256] = 3 MB
  float* fvs  = selT + (size_t)ND * I_DIM;          // [1024, 3072] = 12 MB

  // out is accumulated with atomics -> zero it (async memset is capture-safe)
  hipMemsetAsync(out, 0, (size_t)B_DIM * U_DIM * sizeof(float), stream);

  // K1: 131072 rows / 256 threads
  k_sparsemax<<<(I_DIM * N_DIM) / 256, 256, 0, stream>>>(fsl, selT);

  // K2: 3072 (32x32) tiles, 4 waves/block
  k_gemm1<<<((B_DIM / 32) * (ND / 32)) / 4, 128, 0, stream>>>(x, selT, th, lt, fvs);

  // K3: 64 batch tiles x 16 n-chunks, one wave each
  dim3 g3(B_DIM / 16, 16);
  k_leafgemm<<<g3, 32, 0, stream>>>(fvs, rsp, out);
}